// Arma_14491219656874
// MI455X (gfx1250) — compile-verified
//
#include <hip/hip_runtime.h>

// ---------------------------------------------------------------------------
// ARMA GNN (2 layers, K=2 stacks, T=1, shared weights, eval mode) on gfx1250.
// Layer-1 dense parts (x@Wi1, x@Wr1 for both stacks) use V_WMMA_F32_16X16X4_F32.
// Edge aggregation uses hardware f32 global atomics.
// ---------------------------------------------------------------------------

typedef __attribute__((ext_vector_type(2))) float v2f;
typedef __attribute__((ext_vector_type(8))) float v8f;

#define FIN1   58
#define KPAD   60          // FIN1 padded up to multiple of 4
#define FOUT1  16

// ---------------- degree / norm ----------------

__global__ void zero_f32(float* p, int n) {
    int i = blockIdx.x * blockDim.x + threadIdx.x;
    if (i < n) p[i] = 0.0f;
}

__global__ void degree_kernel(const int* __restrict__ col, float* __restrict__ deg, int E) {
    int e = blockIdx.x * blockDim.x + threadIdx.x;
    if (e < E) unsafeAtomicAdd(&deg[col[e]], 1.0f);
}

__global__ void dis_kernel(float* __restrict__ deg, int N) {
    int n = blockIdx.x * blockDim.x + threadIdx.x;
    if (n >= N) return;
    float d = deg[n];
    deg[n] = (d > 0.0f) ? rsqrtf(fmaxf(d, 1.0f)) : 0.0f;
}

// ---------------- layer-1 GEMMs via WMMA ----------------
// One wave handles a tile of 16 nodes; computes 4 output tiles (Wi k=0/1 -> h1,
// Wr k=0/1 -> acc1 = root + bias). K loop: 15 steps of 16x16x4 f32 WMMA.

__global__ __launch_bounds__(256) void gemm1_wmma(
    const float* __restrict__ x,
    const float* __restrict__ Wi1, const float* __restrict__ Wr1,
    const float* __restrict__ b1,
    float* __restrict__ h1, float* __restrict__ acc1, int N)
{
    __shared__ float lds_w[4 * KPAD * FOUT1];   // 4 matrices, K padded w/ zeros

    const int tid = threadIdx.x;
    for (int idx = tid; idx < 4 * KPAD * FOUT1; idx += 256) {
        int m = idx / (KPAD * FOUT1);
        int r = idx % (KPAD * FOUT1);
        int k = r / FOUT1, o = r % FOUT1;
        float v = 0.0f;
        if (k < FIN1) {
            if (m < 2) v = Wi1[m * FIN1 * FOUT1 + k * FOUT1 + o];
            else       v = Wr1[(m - 2) * FIN1 * FOUT1 + k * FOUT1 + o];
        }
        lds_w[idx] = v;
    }
    __syncthreads();

    const int wave   = tid >> 5;
    const int lane   = tid & 31;
    const int ntiles = N >> 4;                     // N is a multiple of 16
    const int tile   = blockIdx.x * 8 + wave;
    if (tile >= ntiles) return;                    // wave-uniform: EXEC all-1s below

    const int  node_base = tile << 4;
    const int  mrow      = lane & 15;
    const bool hi        = lane >= 16;

    v8f acc[4] = {v8f{}, v8f{}, v8f{}, v8f{}};

    const float* xrow = x + (size_t)(node_base + mrow) * FIN1;

    #pragma unroll
    for (int k0 = 0; k0 < KPAD; k0 += 4) {
        const int kA = k0 + (hi ? 2 : 0);          // K pair this half-wave supplies
        v2f a;
        if (kA < FIN1) {                           // 8-byte aligned (58*n + even k)
            const float2 t = *reinterpret_cast<const float2*>(xrow + kA);
            a.x = t.x; a.y = t.y;
        } else {
            a.x = 0.0f; a.y = 0.0f;                // zero-pad K=58,59
        }
        #pragma unroll
        for (int m = 0; m < 4; ++m) {
            v2f b;
            b.x = lds_w[m * KPAD * FOUT1 + (kA    ) * FOUT1 + mrow];
            b.y = lds_w[m * KPAD * FOUT1 + (kA + 1) * FOUT1 + mrow];
            acc[m] = __builtin_amdgcn_wmma_f32_16x16x4_f32(
                false, a, false, b, (short)0, acc[m], false, false);
        }
    }

    // C/D layout: VGPR v, lanes 0-15 -> row v, lanes 16-31 -> row v+8, col = lane%16
    const int o     = mrow;
    const int rbase = node_base + (hi ? 8 : 0);
    const float bia0 = b1[o], bia1 = b1[FOUT1 + o];
    #pragma unroll
    for (int v = 0; v < 8; ++v) {
        const size_t off = (size_t)(rbase + v) * FOUT1 + o;
        h1[off]                        = acc[0][v];
        h1[(size_t)N * FOUT1 + off]    = acc[1][v];
        acc1[off]                      = acc[2][v] + bia0;
        acc1[(size_t)N * FOUT1 + off]  = acc[3][v] + bia1;
    }
}

// ---------------- layer-1 edge scatter ----------------
// 16 lanes per edge (one per output feature): acc1[k][col] += norm * h1[k][row]

__global__ void scatter1_kernel(const int* __restrict__ ei,
                                const float* __restrict__ dis,
                                const float* __restrict__ h1,
                                float* __restrict__ acc1, int N, int E)
{
    long long t = (long long)blockIdx.x * blockDim.x + threadIdx.x;
    if (t >= (long long)E * FOUT1) return;
    const int e = (int)(t >> 4);
    const int o = (int)(t & 15);
    const int row = ei[e];
    const int col = ei[E + e];
    const float nrm = dis[row] * dis[col];
    const size_t ro = (size_t)row * FOUT1 + o;
    const size_t co = (size_t)col * FOUT1 + o;
    unsafeAtomicAdd(&acc1[co],                    nrm * h1[ro]);
    unsafeAtomicAdd(&acc1[(size_t)N * FOUT1 + co], nrm * h1[(size_t)N * FOUT1 + ro]);
}

// ---------------- relu + mean over stacks (+ wrapper relu, identity here) ----

__global__ void relumean1_kernel(const float* __restrict__ acc1,
                                 float* __restrict__ y, int N)
{
    int i = blockIdx.x * blockDim.x + threadIdx.x;
    if (i >= N * FOUT1) return;
    const float a = fmaxf(acc1[i], 0.0f);
    const float b = fmaxf(acc1[(size_t)N * FOUT1 + i], 0.0f);
    y[i] = 0.5f * (a + b);   // already non-negative; outer relu is identity
}

// ---------------- layer-2 dense (F 16 -> 1, 2 stacks) ----------------

__global__ void gemm2_kernel(const float* __restrict__ y,
                             const float* __restrict__ Wi2,
                             const float* __restrict__ Wr2,
                             const float* __restrict__ b2,
                             float* __restrict__ h2, float* __restrict__ acc2, int N)
{
    int n = blockIdx.x * blockDim.x + threadIdx.x;
    if (n >= N) return;
    const float* yr = y + (size_t)n * FOUT1;
    float di0 = 0.f, di1 = 0.f, dr0 = 0.f, dr1 = 0.f;
    #pragma unroll
    for (int f = 0; f < FOUT1; ++f) {
        const float v = yr[f];
        di0 += v * Wi2[f];
        di1 += v * Wi2[FOUT1 + f];
        dr0 += v * Wr2[f];
        dr1 += v * Wr2[FOUT1 + f];
    }
    h2[n]       = di0;
    h2[N + n]   = di1;
    acc2[n]     = dr0 + b2[0];
    acc2[N + n] = dr1 + b2[1];
}

// ---------------- layer-2 edge scatter ----------------

__global__ void scatter2_kernel(const int* __restrict__ ei,
                                const float* __restrict__ dis,
                                const float* __restrict__ h2,
                                float* __restrict__ acc2, int N, int E)
{
    int e = blockIdx.x * blockDim.x + threadIdx.x;
    if (e >= E) return;
    const int row = ei[e];
    const int col = ei[E + e];
    const float nrm = dis[row] * dis[col];
    unsafeAtomicAdd(&acc2[col],     nrm * h2[row]);
    unsafeAtomicAdd(&acc2[N + col], nrm * h2[N + row]);
}

// ---------------- final relu + mean ----------------

__global__ void final_kernel(const float* __restrict__ acc2,
                             float* __restrict__ out, int N)
{
    int n = blockIdx.x * blockDim.x + threadIdx.x;
    if (n >= N) return;
    out[n] = 0.5f * (fmaxf(acc2[n], 0.0f) + fmaxf(acc2[N + n], 0.0f));
}

// ---------------------------------------------------------------------------

extern "C" void kernel_launch(void* const* d_in, const int* in_sizes, int n_in,
                              void* d_out, int out_size, void* d_ws, size_t ws_size,
                              hipStream_t stream)
{
    const float* x   = (const float*)d_in[0];
    const int*   ei  = (const int*)  d_in[1];
    const float* Wi1 = (const float*)d_in[2];
    const float* Wr1 = (const float*)d_in[3];
    const float* b1  = (const float*)d_in[4];
    const float* Wi2 = (const float*)d_in[5];
    const float* Wr2 = (const float*)d_in[6];
    const float* b2  = (const float*)d_in[7];
    float* out = (float*)d_out;

    const int N = in_sizes[0] / FIN1;   // 100000
    const int E = in_sizes[1] / 2;      // 1600000

    // Workspace layout (floats); N is a multiple of 16 so everything stays aligned.
    float* ws   = (float*)d_ws;
    float* dis  = ws;                                  // [N]        (deg -> dis in place)
    float* h1   = dis  + (size_t)N;                    // [2, N, 16]
    float* acc1 = h1   + (size_t)2 * N * FOUT1;        // [2, N, 16]
    float* y    = acc1 + (size_t)2 * N * FOUT1;        // [N, 16]
    float* h2   = y    + (size_t)N * FOUT1;            // [2, N]
    float* acc2 = h2   + (size_t)2 * N;                // [2, N]

    const int B = 256;

    // 1) degrees (must re-zero every call: harness does not re-init scratch)
    zero_f32<<<(N + B - 1) / B, B, 0, stream>>>(dis, N);
    degree_kernel<<<(E + B - 1) / B, B, 0, stream>>>(ei + E, dis, E);
    dis_kernel<<<(N + B - 1) / B, B, 0, stream>>>(dis, N);

    // 2) layer 1 dense: WMMA, 8 waves/block, 16-node tile per wave
    const int ntiles = N / 16;
    gemm1_wmma<<<(ntiles + 7) / 8, B, 0, stream>>>(x, Wi1, Wr1, b1, h1, acc1, N);

    // 3) layer 1 aggregation (16 lanes per edge)
    const long long s1 = (long long)E * FOUT1;
    scatter1_kernel<<<(int)((s1 + B - 1) / B), B, 0, stream>>>(ei, dis, h1, acc1, N, E);

    // 4) relu + mean over stacks
    relumean1_kernel<<<(N * FOUT1 + B - 1) / B, B, 0, stream>>>(acc1, y, N);

    // 5) layer 2 dense
    gemm2_kernel<<<(N + B - 1) / B, B, 0, stream>>>(y, Wi2, Wr2, b2, h2, acc2, N);

    // 6) layer 2 aggregation
    scatter2_kernel<<<(E + B - 1) / B, B, 0, stream>>>(ei, dis, h2, acc2, N, E);

    // 7) final relu + mean -> [N, 1]
    final_kernel<<<(N + B - 1) / B, B, 0, stream>>>(acc2, out, N);
}